// GatingLayer_47991964566217
// MI455X (gfx1250) — compile-verified
//
#include <hip/hip_runtime.h>
#include <stdint.h>

// -------------------------------------------------------------------------
// Logistic-map chaos gate on MI455X (gfx1250).
//
// Latency-bound serial recurrence x_{n+1} = r*x*(1-x): one wave32 runs the
// chain (all lanes redundantly compute the same value; dataflow depth 2 per
// step, and the compiler VOPD-pairs the independent (1-x) and (r*x) ops).
// Lane (i%32) captures step i via one v_cndmask off hoisted SGPR lane masks.
// Every 32 steps -> one coalesced ds_store_b32 row into an LDS staging
// buffer; every CHUNK elements -> one TENSOR_STORE_FROM_LDS DMA pushes the
// buffer to HBM while the wave keeps iterating (double buffered via
// s_wait_tensorcnt). The VALU pipe never issues a global store.
// -------------------------------------------------------------------------

typedef unsigned int v4u __attribute__((ext_vector_type(4)));
typedef int          v4i __attribute__((ext_vector_type(4)));
typedef int          v8i __attribute__((ext_vector_type(8)));

#define CHUNK 4096   // floats per TDM transfer (16 KB)
#define BATCH 32     // recurrence steps per LDS row (one value per lane)

// Build a D# for a 1-D tile of `nelem` 4-byte elements and issue the DMA.
// Group0: count=1 | lds_addr | global_addr[56:0] | type=2 ("image").
// Group1: data_size=2 (4B), tensor_dim0 = tile_dim0 = nelem, tensor_dim1 = 1,
//         tile_dim1 = 1, tensor_dim0_stride = nelem, no padding/iterate.
__device__ __forceinline__ void tdm_store_chunk(uint32_t lds_byte_addr,
                                                uint64_t gaddr,
                                                uint32_t nelem) {
  v4u g0;
  g0[0] = 1u;                                              // count=1, user desc
  g0[1] = lds_byte_addr;                                   // lds_addr (bytes)
  g0[2] = (uint32_t)(gaddr & 0xFFFFFFFFull);               // global_addr lo
  g0[3] = (uint32_t)((gaddr >> 32) & 0x1FFFFFFull)         // global_addr hi
        | (2u << 30);                                      // type = 2

  v8i g1;
  g1[0] = (int)(2u << 16);                                 // data_size=4B, mask=0
  g1[1] = (int)((nelem & 0xFFFFu) << 16);                  // tensor_dim0[15:0]
  g1[2] = (int)(((nelem >> 16) & 0xFFFFu) | (1u << 16));   // dim0 hi | tensor_dim1=1
  g1[3] = (int)((nelem & 0xFFFFu) << 16);                  // dim1 hi=0 | tile_dim0
  g1[4] = 1;                                               // tile_dim1=1, tile_dim2=0
  g1[5] = (int)nelem;                                      // tensor_dim0_stride lo
  g1[6] = 0;                                               // stride hi | dim1_stride lo
  g1[7] = 0;                                               // tensor_dim1_stride hi

  v4i z = {0, 0, 0, 0};
#if defined(__clang_major__) && (__clang_major__ >= 23)
  v8i z8 = {0, 0, 0, 0, 0, 0, 0, 0};
  __builtin_amdgcn_tensor_store_from_lds(g0, g1, z, z, z8, 0);
#else
  __builtin_amdgcn_tensor_store_from_lds(g0, g1, z, z, 0);
#endif
}

// One faithful logistic-map step + lane capture. Explicit op order
// (sub / mul / mul via *_rn intrinsics) forbids FMA contraction: the map is
// chaotic, so any re-association diverges from the reference.
// The (lane == J) compares are loop-invariant -> hoisted to 32 SGPR masks;
// the capture is a single v_cndmask per step, off the critical path.
#define STEP(J)                                   \
  do {                                            \
    float s_ = __fsub_rn(1.0f, x);                \
    float t_ = __fmul_rn(r, x);                   \
    x        = __fmul_rn(t_, s_);                 \
    cap = (lane == (unsigned)(J)) ? x : cap;      \
  } while (0)

__global__ void __launch_bounds__(32, 1)
chaos_gate_kernel(const float* __restrict__ x0p,
                  const float* __restrict__ rp,
                  float* __restrict__ out, int n) {
  __shared__ float stage[2][CHUNK];

  const unsigned lane = threadIdx.x & 31u;
  float x       = x0p[0];
  const float r = rp[0];

  // LDS byte offset of the staging buffer (flat addr low 32 bits == LDS offset).
  const uint32_t lds_base = (uint32_t)(uintptr_t)(&stage[0][0]);

  const int nchunks = n / CHUNK;
  int i = 0;

  for (int c = 0; c < nchunks; ++c) {
    const int buf = c & 1;

    // The TDM op that last read this buffer was issued at chunk c-2; TDM ops
    // complete in order, so <=1 outstanding guarantees the buffer is free.
    __builtin_amdgcn_s_wait_tensorcnt(1);

    float* dstp = &stage[buf][0] + lane;
    for (int b = 0; b < CHUNK / BATCH; ++b) {
      float cap = 0.0f;
      STEP(0);  STEP(1);  STEP(2);  STEP(3);
      STEP(4);  STEP(5);  STEP(6);  STEP(7);
      STEP(8);  STEP(9);  STEP(10); STEP(11);
      STEP(12); STEP(13); STEP(14); STEP(15);
      STEP(16); STEP(17); STEP(18); STEP(19);
      STEP(20); STEP(21); STEP(22); STEP(23);
      STEP(24); STEP(25); STEP(26); STEP(27);
      STEP(28); STEP(29); STEP(30); STEP(31);
      *dstp = cap;            // one ds_store_b32 per 32 steps, conflict-free
      dstp += BATCH;
    }
    i += CHUNK;

    // All ds_stores must land in LDS before the TDM reads it.
    asm volatile("s_wait_dscnt 0" ::: "memory");

    tdm_store_chunk(lds_base + (uint32_t)buf * (CHUNK * 4u),
                    (uint64_t)(uintptr_t)(out + (size_t)c * CHUNK),
                    (uint32_t)CHUNK);
  }

  // Tail for n not divisible by CHUNK (n = 4M is divisible; defensive only).
  for (; i < n; ++i) {
    float s = __fsub_rn(1.0f, x);
    float t = __fmul_rn(r, x);
    x       = __fmul_rn(t, s);
    if (lane == 0u) out[i] = x;
  }

  // Drain outstanding tensor DMAs before wave exit.
  __builtin_amdgcn_s_wait_tensorcnt(0);
}

extern "C" void kernel_launch(void* const* d_in, const int* in_sizes, int n_in,
                              void* d_out, int out_size, void* d_ws, size_t ws_size,
                              hipStream_t stream) {
  (void)in_sizes; (void)n_in; (void)d_ws; (void)ws_size;
  // setup_inputs order: d_in[0]=length (int), d_in[1]=x0 (f32), d_in[2]=r (f32)
  const float* x0 = (const float*)d_in[1];
  const float* r  = (const float*)d_in[2];
  float* out      = (float*)d_out;

  // Single wave: the recurrence is a serial dependency chain; extra waves
  // cannot help. All output data movement is offloaded to the Tensor Data
  // Mover so the VALU pipe carries only the recurrence.
  chaos_gate_kernel<<<1, 32, 0, stream>>>(x0, r, out, out_size);
}